// CausalGraphGAT_59768764891880
// MI455X (gfx1250) — compile-verified
//
#include <hip/hip_runtime.h>
#include <math.h>

// ---------------------------------------------------------------------------
// CausalGraphGAT forward for MI455X (gfx1250, wave32).
//   - Dense h @ W via V_WMMA_F32_16X16X4_F32 (exact fp32 matrix path),
//     A tiles staged to LDS with GLOBAL_LOAD_ASYNC_TO_LDS_B128 (ASYNCcnt),
//     two row-tiles per wave sharing B fragments (64 WMMA / wave).
//   - Edge softmax/aggregation via L2-resident atomics; topology decoded once
//     to int32; float4 message scatter.
// Input order (insertion order of setup_inputs / params dicts):
//   0:x 1:edge_index(i64) 2:batch(i64)
//   3..6 L0{W,att_src,att_dst,bias} 7..10 L1{...} 11..14 L2{...}
//   15:cls_W 16:cls_b 17:reg_W 18:reg_b 19:g1_W 20:g1_b 21:g2_W 22:g2_b
// Output: [emb N*128 | node_cls N | node_reg N | graph_pred 50]
// ---------------------------------------------------------------------------

#define FW   128
#define NGR  50
#define SLOPE 0.2f

typedef __attribute__((ext_vector_type(2))) float v2f;
typedef __attribute__((ext_vector_type(8))) float v8f;

// Order-preserving float<->int map: float segment-max uses native
// GLOBAL_ATOMIC_MAX_I32 instead of a CAS loop.
__device__ __forceinline__ int fmap_ord(float f) {
  int i = __float_as_int(f);
  return i >= 0 ? i : (i ^ 0x7FFFFFFF);
}
__device__ __forceinline__ float funmap_ord(int i) {
  return __int_as_float(i >= 0 ? i : (i ^ 0x7FFFFFFF));
}

// ---------------------------------------------------------------------------
// GEMM: Y[nrows,128] = X[nrows,128] @ W[128,128], f32 WMMA 16x16x4.
// Block = 32-row strip; 8 waves each own one 16-col tile and TWO 16-row
// tiles (acc0/acc1) that share B fragments. A staged in LDS with stride 132:
// lane l -> banks {4l,4l+1}, lane 16+l -> {4l+2,4l+3} => conflict-free
// ds_load_b64 A-fragment reads. B loads use one base + immediate offsets.
// ---------------------------------------------------------------------------
__global__ void gemm128_wmma(const float* __restrict__ X,
                             const float* __restrict__ W,
                             float* __restrict__ Y, int nrows) {
  constexpr int LDA = 132;
  __shared__ float lA[32 * LDA];
  const int r0  = blockIdx.x * 32;
  const int tid = threadIdx.x;

  // Async copy of the 32x128 A tile (16 KB) straight into LDS.
  // GVS mode: saddr = X base, vaddr = 32-bit byte offset, vdst = LDS offset.
  // Low 32 bits of the flat shared-pointer are the LDS byte offset.
  const unsigned ldsbase = (unsigned)(uintptr_t)(void*)lA;
#pragma unroll
  for (int j = 0; j < 4; ++j) {
    int cc = tid + j * 256;            // 16B chunk id, 1024 chunks total
    int r = cc >> 5;                   // tile row (0..31)
    int k = (cc & 31) << 2;            // starting float within row
    int row = r0 + r;
    if (row >= nrows) row = nrows - 1; // clamp, never OOB
    unsigned ldsoff = ldsbase + (unsigned)((r * LDA + k) * 4);
    unsigned goff   = (unsigned)(((size_t)row * FW + k) * 4);
    asm volatile("global_load_async_to_lds_b128 %0, %1, %2"
                 :: "v"(ldsoff), "v"(goff), "s"(X) : "memory");
  }
  asm volatile("s_wait_asynccnt 0x0" ::: "memory");
  __syncthreads();

  const int wave = tid >> 5;
  const int lane = tid & 31;
  const int hi   = lane >> 4;          // 0: K={0,1}, 1: K={2,3} per k-step
  const int l16  = lane & 15;
  const int n0   = wave * 16;

  const float* lAp = lA + l16 * LDA + 2 * hi;              // 8B aligned
  const float* wp  = W + (size_t)(2 * hi) * FW + n0 + l16; // + imm k0*512B

  v8f acc0 = {}, acc1 = {};
#pragma unroll
  for (int k0 = 0; k0 < FW; k0 += 4) {
    v2f a0 = *(const v2f*)(lAp + k0);                 // rows r0..r0+15
    v2f a1 = *(const v2f*)(lAp + 16 * LDA + k0);      // rows r0+16..r0+31
    v2f b;
    b.x = wp[k0 * FW];                                // W[k0+2hi,   n0+l16]
    b.y = wp[k0 * FW + FW];                           // W[k0+2hi+1, n0+l16]
    acc0 = __builtin_amdgcn_wmma_f32_16x16x4_f32(
        false, a0, false, b, (short)0, acc0, false, false);
    acc1 = __builtin_amdgcn_wmma_f32_16x16x4_f32(
        false, a1, false, b, (short)0, acc1, false, false);
  }

  // C/D layout: vgpr i -> M = i + 8*hi, N = l16. Uniform tail guards only.
  if (r0 + 16 <= nrows) {
    float* yp = Y + (size_t)(r0 + 8 * hi) * FW + n0 + l16;
#pragma unroll
    for (int i = 0; i < 8; ++i) yp[(size_t)i * FW] = acc0[i];
  } else {
#pragma unroll
    for (int i = 0; i < 8; ++i) {
      int m = i + 8 * hi;
      if (r0 + m < nrows) Y[(size_t)(r0 + m) * FW + n0 + l16] = acc0[i];
    }
  }
  if (r0 + 32 <= nrows) {
    float* yp = Y + (size_t)(r0 + 16 + 8 * hi) * FW + n0 + l16;
#pragma unroll
    for (int i = 0; i < 8; ++i) yp[(size_t)i * FW] = acc1[i];
  } else {
#pragma unroll
    for (int i = 0; i < 8; ++i) {
      int m = 16 + i + 8 * hi;
      if (r0 + m < nrows) Y[(size_t)(r0 + m) * FW + n0 + l16] = acc1[i];
    }
  }
}

// Decode int64 edge_index (+ appended self loops) once into int32 arrays.
__global__ void decode_edges_k(const long long* __restrict__ ei, int E, int n,
                               int* __restrict__ src32, int* __restrict__ dst32) {
  int e = blockIdx.x * 256 + threadIdx.x;
  if (e >= E + n) return;
  if (e < E) {
    src32[e] = (int)ei[e];
    dst32[e] = (int)ei[(size_t)E + e];
  } else {
    src32[e] = dst32[e] = e - E;       // self loop
  }
}

// a_s[n,h] = sum_d xl[n,h,d]*att_src[h,d]; a_d likewise.
template <int H, int D>
__global__ void node_att_k(const float* __restrict__ xl,
                           const float* __restrict__ att_s,
                           const float* __restrict__ att_d,
                           float* __restrict__ a_s, float* __restrict__ a_d,
                           int n) {
  int idx = blockIdx.x * 256 + threadIdx.x;
  if (idx >= n * H) return;
  int node = idx / H, h = idx - node * H;
  const float* row = xl + (size_t)node * FW + h * D;
  float s = 0.f, d = 0.f;
#pragma unroll 8
  for (int k = 0; k < D; ++k) {
    float v = row[k];
    s += v * att_s[h * D + k];
    d += v * att_d[h * D + k];
  }
  a_s[idx] = s;
  a_d[idx] = d;
}

// acc rows <- bias (bias folded into scatter), m <- -inf, ssum <- 0.
__global__ void init_layer_k(float* __restrict__ acc, const float* __restrict__ bias,
                             int* __restrict__ m, float* __restrict__ ssum,
                             int n, int H) {
  int i = blockIdx.x * 256 + threadIdx.x;
  if (i < n * FW) acc[i] = bias[i & 127];
  if (i < n * H) { m[i] = fmap_ord(-INFINITY); ssum[i] = 0.0f; }
}

template <int H>
__global__ void edge_max_k(const int* __restrict__ src32, const int* __restrict__ dst32,
                           int E2, const float* __restrict__ a_s,
                           const float* __restrict__ a_d, int* __restrict__ m) {
  int idx = blockIdx.x * 256 + threadIdx.x;
  if (idx >= E2 * H) return;
  int e = idx / H, h = idx & (H - 1);
  int src = src32[e], dst = dst32[e];
  float v = a_s[src * H + h] + a_d[dst * H + h];
  v = v > 0.f ? v : SLOPE * v;                  // leaky_relu
  atomicMax(&m[dst * H + h], fmap_ord(v));      // GLOBAL_ATOMIC_MAX_I32
}

template <int H>
__global__ void edge_expsum_k(const int* __restrict__ src32, const int* __restrict__ dst32,
                              int E2, const float* __restrict__ a_s,
                              const float* __restrict__ a_d, const int* __restrict__ m,
                              float* __restrict__ ssum, float* __restrict__ eexp) {
  int idx = blockIdx.x * 256 + threadIdx.x;
  if (idx >= E2 * H) return;
  int e = idx / H, h = idx & (H - 1);
  int src = src32[e], dst = dst32[e];
  float v = a_s[src * H + h] + a_d[dst * H + h];
  v = v > 0.f ? v : SLOPE * v;
  float w = expf(v - funmap_ord(m[dst * H + h]));
  eexp[idx] = w;
  atomicAdd(&ssum[dst * H + h], w);             // GLOBAL_ATOMIC_ADD_F32
}

// eexp -> alpha in place (removes the ssum gather from the hot scatter).
template <int H>
__global__ void edge_alpha_k(const int* __restrict__ dst32, int E2,
                             const float* __restrict__ ssum, float* __restrict__ eexp) {
  int idx = blockIdx.x * 256 + threadIdx.x;
  if (idx >= E2 * H) return;
  int e = idx / H, h = idx & (H - 1);
  int dst = dst32[e];
  eexp[idx] = eexp[idx] / (ssum[dst * H + h] + 1e-16f);
}

// acc[dst, f0..f0+3] += alpha(e,h) * xl[src, f0..f0+3]; 32 threads / edge,
// float4 loads, coalesced f32 atomics. e is wave-uniform; h/alpha are too only when D = 128.
template <int H, int D>
__global__ void edge_msg_k(const int* __restrict__ src32, const int* __restrict__ dst32,
                           int E2, const float* __restrict__ xl,
                           const float* __restrict__ alpha, float* __restrict__ acc) {
  int idx = blockIdx.x * 256 + threadIdx.x;     // E2*32 = 33.6M < 2^31
  if (idx >= E2 * 32) return;
  int e  = idx >> 5;
  int f0 = (idx & 31) << 2;
  int h  = f0 / D;                              // compile-time D -> shift
  int src = src32[e], dst = dst32[e];
  float a = alpha[e * H + h];
  const float4 v = *(const float4*)(xl + (size_t)src * FW + f0);
  float* ap = acc + (size_t)dst * FW + f0;
  atomicAdd(ap + 0, a * v.x);
  atomicAdd(ap + 1, a * v.y);
  atomicAdd(ap + 2, a * v.z);
  atomicAdd(ap + 3, a * v.w);
}

__global__ void elu_k(float* __restrict__ x, int count) {
  int i = blockIdx.x * 256 + threadIdx.x;
  if (i >= count) return;
  float v = x[i];
  x[i] = v > 0.f ? v : (expf(v) - 1.0f);
}

// One wave per node: 128-wide dots with cls_W and reg_W.
__global__ void node_heads_k(const float* __restrict__ emb,
                             const float* __restrict__ cls_W, const float* __restrict__ cls_b,
                             const float* __restrict__ reg_W, const float* __restrict__ reg_b,
                             float* __restrict__ node_cls, float* __restrict__ node_reg,
                             int n) {
  int wave = threadIdx.x >> 5, lane = threadIdx.x & 31;
  int node = blockIdx.x * 8 + wave;
  if (node >= n) return;
  const float* row = emb + (size_t)node * FW;
  float c = 0.f, r = 0.f;
#pragma unroll
  for (int j = 0; j < 4; ++j) {
    int k = lane + 32 * j;
    float v = row[k];
    c += v * cls_W[k];
    r += v * reg_W[k];
  }
#pragma unroll
  for (int off = 16; off > 0; off >>= 1) {      // wave32 shuffle reduce
    c += __shfl_down(c, off, 32);
    r += __shfl_down(r, off, 32);
  }
  if (lane == 0) {
    node_cls[node] = 1.0f / (1.0f + expf(-(c + cls_b[0])));
    node_reg[node] = r + reg_b[0];
  }
}

__global__ void init_pool_k(float* __restrict__ g_sum, int* __restrict__ g_max,
                            float* __restrict__ counts) {
  int i = blockIdx.x * 256 + threadIdx.x;
  if (i < NGR * FW) { g_sum[i] = 0.0f; g_max[i] = fmap_ord(-INFINITY); }
  if (i < NGR) counts[i] = 0.0f;
}

__global__ void pool_accum_k(const float* __restrict__ emb, const long long* __restrict__ batch,
                             float* __restrict__ g_sum, int* __restrict__ g_max,
                             float* __restrict__ counts, int n) {
  int idx = blockIdx.x * 256 + threadIdx.x;
  if (idx >= n * FW) return;
  int node = idx >> 7, f = idx & 127;
  int g = (int)batch[node];
  float v = emb[idx];
  atomicAdd(&g_sum[g * FW + f], v);
  atomicMax(&g_max[g * FW + f], fmap_ord(v));
  if (f == 0) atomicAdd(&counts[g], 1.0f);
}

// One block/graph: gf=[mean|max](256) -> gh=relu(gf@g1_W+b)(128) -> sigmoid dot.
__global__ void graph_mlp_k(const float* __restrict__ g_sum, const int* __restrict__ g_max,
                            const float* __restrict__ counts,
                            const float* __restrict__ g1_W, const float* __restrict__ g1_b,
                            const float* __restrict__ g2_W, const float* __restrict__ g2_b,
                            float* __restrict__ graph_pred) {
  __shared__ float gf[2 * FW];
  __shared__ float red[FW];
  int g = blockIdx.x, t = threadIdx.x;          // 128 threads
  float c = counts[g];
  float inv = 1.0f / fmaxf(c, 1.0f);
  gf[t]      = g_sum[g * FW + t] * inv;
  gf[FW + t] = (c > 0.f) ? funmap_ord(g_max[g * FW + t]) : 0.0f;
  __syncthreads();
  float acc = g1_b[t];
  for (int k = 0; k < 2 * FW; ++k) acc += gf[k] * g1_W[k * FW + t];
  red[t] = fmaxf(acc, 0.0f) * g2_W[t];
  __syncthreads();
  for (int s = 64; s > 0; s >>= 1) {
    if (t < s) red[t] += red[t + s];
    __syncthreads();
  }
  if (t == 0) graph_pred[g] = 1.0f / (1.0f + expf(-(red[0] + g2_b[0])));
}

// ---------------------------------------------------------------------------

static inline int cdiv(long long a, int b) { return (int)((a + b - 1) / b); }

template <int H, int D>
static void run_layer(const float* Xin, const float* W, const float* att_s,
                      const float* att_d, const float* bias, float* Aout,
                      float* xl, float* a_s, float* a_d, int* segm, float* ssum,
                      float* eexp, const int* src32, const int* dst32,
                      int n, int E2, bool elu, hipStream_t stream) {
  gemm128_wmma<<<cdiv(n, 32), 256, 0, stream>>>(Xin, W, xl, n);
  node_att_k<H, D><<<cdiv((long long)n * H, 256), 256, 0, stream>>>(xl, att_s, att_d, a_s, a_d, n);
  init_layer_k<<<cdiv((long long)n * FW, 256), 256, 0, stream>>>(Aout, bias, segm, ssum, n, H);
  edge_max_k<H><<<cdiv((long long)E2 * H, 256), 256, 0, stream>>>(src32, dst32, E2, a_s, a_d, segm);
  edge_expsum_k<H><<<cdiv((long long)E2 * H, 256), 256, 0, stream>>>(src32, dst32, E2, a_s, a_d,
                                                                     segm, ssum, eexp);
  edge_alpha_k<H><<<cdiv((long long)E2 * H, 256), 256, 0, stream>>>(dst32, E2, ssum, eexp);
  edge_msg_k<H, D><<<cdiv((long long)E2 * 32, 256), 256, 0, stream>>>(src32, dst32, E2, xl, eexp, Aout);
  if (elu) elu_k<<<cdiv((long long)n * FW, 256), 256, 0, stream>>>(Aout, n * FW);
}

extern "C" void kernel_launch(void* const* d_in, const int* in_sizes, int n_in,
                              void* d_out, int out_size, void* d_ws, size_t ws_size,
                              hipStream_t stream) {
  const float*     x     = (const float*)d_in[0];
  const long long* ei    = (const long long*)d_in[1];   // [2,E] int64
  const long long* batch = (const long long*)d_in[2];   // [N]   int64
  const float* LW[3]  = {(const float*)d_in[3],  (const float*)d_in[7],  (const float*)d_in[11]};
  const float* LAS[3] = {(const float*)d_in[4],  (const float*)d_in[8],  (const float*)d_in[12]};
  const float* LAD[3] = {(const float*)d_in[5],  (const float*)d_in[9],  (const float*)d_in[13]};
  const float* LB[3]  = {(const float*)d_in[6],  (const float*)d_in[10], (const float*)d_in[14]};
  const float* cls_W = (const float*)d_in[15];
  const float* cls_b = (const float*)d_in[16];
  const float* reg_W = (const float*)d_in[17];
  const float* reg_b = (const float*)d_in[18];
  const float* g1_W  = (const float*)d_in[19];
  const float* g1_b  = (const float*)d_in[20];
  const float* g2_W  = (const float*)d_in[21];
  const float* g2_b  = (const float*)d_in[22];

  const int n  = in_sizes[0] / FW;     // 50000
  const int E  = in_sizes[1] / 2;      // 1000000
  const int E2 = E + n;

  float* out        = (float*)d_out;
  float* emb        = out;                               // [n,128]
  float* node_cls   = out + (size_t)n * FW;
  float* node_reg   = node_cls + n;
  float* graph_pred = node_reg + n;

  // workspace carve (4-byte elems)
  float* ws = (float*)d_ws;
  size_t o = 0;
  float* xl    = ws + o; o += (size_t)n * FW;
  float* hA    = ws + o; o += (size_t)n * FW;
  float* hB    = ws + o; o += (size_t)n * FW;
  float* a_s   = ws + o; o += (size_t)n * 4;
  float* a_d   = ws + o; o += (size_t)n * 4;
  int*   segm  = (int*)(ws + o); o += (size_t)n * 4;
  float* ssum  = ws + o; o += (size_t)n * 4;
  float* eexp  = ws + o; o += (size_t)E2 * 4;
  int*   src32 = (int*)(ws + o); o += (size_t)E2;
  int*   dst32 = (int*)(ws + o); o += (size_t)E2;
  float* g_sum = ws + o; o += (size_t)NGR * FW;
  int*   g_max = (int*)(ws + o); o += (size_t)NGR * FW;
  float* counts = ws + o; o += NGR;
  (void)ws_size; (void)n_in; (void)out_size;

  // Topology is layer-invariant: decode once.
  decode_edges_k<<<cdiv(E2, 256), 256, 0, stream>>>(ei, E, n, src32, dst32);

  run_layer<4, 32>(x,  LW[0], LAS[0], LAD[0], LB[0], hA,  xl, a_s, a_d, segm, ssum,
                   eexp, src32, dst32, n, E2, true,  stream);
  run_layer<4, 32>(hA, LW[1], LAS[1], LAD[1], LB[1], hB,  xl, a_s, a_d, segm, ssum,
                   eexp, src32, dst32, n, E2, true,  stream);
  run_layer<1, 128>(hB, LW[2], LAS[2], LAD[2], LB[2], emb, xl, a_s, a_d, segm, ssum,
                    eexp, src32, dst32, n, E2, false, stream);

  node_heads_k<<<cdiv(n, 8), 256, 0, stream>>>(emb, cls_W, cls_b, reg_W, reg_b,
                                               node_cls, node_reg, n);
  init_pool_k<<<cdiv(NGR * FW, 256), 256, 0, stream>>>(g_sum, g_max, counts);
  pool_accum_k<<<cdiv((long long)n * FW, 256), 256, 0, stream>>>(emb, batch, g_sum, g_max, counts, n);
  graph_mlp_k<<<NGR, FW, 0, stream>>>(g_sum, g_max, counts, g1_W, g1_b, g2_W, g2_b, graph_pred);
}